// GMMActivation2DFullyLearnable_23802708755043
// MI455X (gfx1250) — compile-verified
//
#include <hip/hip_runtime.h>
#include <math.h>

// GMM 2D activation: out = x * ( sum_k exp(-0.5 * diff_k^T Sigma_k^{-1} diff_k) * w[d2,k,:] )
// Memory-bound streaming kernel (128 MB traffic, ~5.5us HBM floor @ 23.3 TB/s).
// Strategy: B128 coalesced loads/stores, guaranteed packed-fp32 VOP3P math
// (v_pk_add/mul/fma_f32) across the two (x0,x1) pairs of each float4,
// v_exp_f32 on the trans pipe, column-stationary weights in VGPRs,
// 8 consecutive rows per thread so all addresses fold into 24-bit ioffsets.

typedef __attribute__((ext_vector_type(2))) float v2f;
typedef __attribute__((ext_vector_type(4))) float v4f;

#define D_DIM 2048
#define KCOMP 4
#define COLS (D_DIM / 4)      // 512 float4 columns per (b,s) row
#define COLS_SHIFT 9          // log2(COLS)
#define IPT 8                 // consecutive rows per thread

// ---- packed fp32 VOP3P helpers (CDNA5 V_PK_*_F32); 64-bit aligned VGPR pairs.
__device__ __forceinline__ v2f pk_add(v2f a, v2f b) {
    v2f d; asm("v_pk_add_f32 %0, %1, %2" : "=v"(d) : "v"(a), "v"(b)); return d;
}
__device__ __forceinline__ v2f pk_mul(v2f a, v2f b) {
    v2f d; asm("v_pk_mul_f32 %0, %1, %2" : "=v"(d) : "v"(a), "v"(b)); return d;
}
__device__ __forceinline__ v2f pk_fma(v2f a, v2f b, v2f c) {
    v2f d; asm("v_pk_fma_f32 %0, %1, %2, %3" : "=v"(d) : "v"(a), "v"(b), "v"(c)); return d;
}

__device__ __forceinline__ float fast_exp2(float v) {
#if defined(__HIP_DEVICE_COMPILE__) && __has_builtin(__builtin_amdgcn_exp2f)
    return __builtin_amdgcn_exp2f(v);   // v_exp_f32
#else
    return exp2f(v);
#endif
}

__global__ __launch_bounds__(256) void gmm2d_act_kernel(
    const float* __restrict__ x,
    const float* __restrict__ means,      // (K,2)
    const float* __restrict__ icov,       // (K,2,2)
    const float* __restrict__ wts,        // (D/2,K,2) flat, stride 8 per d2
    float* __restrict__ out,
    int rows)                             // B*S
{
    const int tid  = blockIdx.x * blockDim.x + threadIdx.x;
    const int col  = tid & (COLS - 1);                    // fixed float4 column
    const int row0 = (tid >> COLS_SHIFT) * IPT;           // 8 consecutive rows

    // ---- uniform coefficients on SALU; fold -0.5*log2(e) into Sigma^-1 and
    // pre-negate means so the diff becomes a pk_add.
    // pdf = exp2( A*d0^2 + B*d0*d1 + C*d1^2 ),  d = x + (-m)
    const float s = -0.72134752044448170368f;             // -0.5 * log2(e)
    v2f NM0[KCOMP], NM1[KCOMP], Ak[KCOMP], Bk[KCOMP], Ck[KCOMP];
#pragma unroll
    for (int k = 0; k < KCOMP; ++k) {
        const float nm0 = -means[2 * k + 0];
        const float nm1 = -means[2 * k + 1];
        const float a   = s * icov[4 * k + 0];
        const float b   = s * (icov[4 * k + 1] + icov[4 * k + 2]);
        const float c   = s * icov[4 * k + 3];
        NM0[k] = (v2f){nm0, nm0};
        NM1[k] = (v2f){nm1, nm1};
        Ak[k]  = (v2f){a, a};
        Bk[k]  = (v2f){b, b};
        Ck[k]  = (v2f){c, c};
    }

    // ---- column-stationary weights: 16 floats for this thread's two pairs,
    // repacked across the pair dimension so packed-fp32 FMAs apply directly.
    const v4f* wv = (const v4f*)wts + (size_t)col * 4;    // pairs 2*col, 2*col+1
    const v4f wa0 = wv[0];   // pair A: k0c0 k0c1 k1c0 k1c1
    const v4f wa1 = wv[1];   // pair A: k2c0 k2c1 k3c0 k3c1
    const v4f wb0 = wv[2];   // pair B: k0c0 k0c1 k1c0 k1c1
    const v4f wb1 = wv[3];   // pair B: k2c0 k2c1 k3c0 k3c1
    v2f W0[KCOMP], W1[KCOMP];
    W0[0] = (v2f){wa0.x, wb0.x};  W1[0] = (v2f){wa0.y, wb0.y};
    W0[1] = (v2f){wa0.z, wb0.z};  W1[1] = (v2f){wa0.w, wb0.w};
    W0[2] = (v2f){wa1.x, wb1.x};  W1[2] = (v2f){wa1.y, wb1.y};
    W0[3] = (v2f){wa1.z, wb1.z};  W1[3] = (v2f){wa1.w, wb1.w};

    auto process = [&](v4f xv) -> v4f {
        const v2f x0 = (v2f){xv.x, xv.z};   // even elements of both pairs
        const v2f x1 = (v2f){xv.y, xv.w};   // odd elements of both pairs
        v2f r0 = (v2f){0.0f, 0.0f};
        v2f r1 = (v2f){0.0f, 0.0f};
#pragma unroll
        for (int k = 0; k < KCOMP; ++k) {
            const v2f d0 = pk_add(x0, NM0[k]);
            const v2f d1 = pk_add(x1, NM1[k]);
            v2f q = pk_mul(Ck[k], pk_mul(d1, d1));
            q = pk_fma(pk_mul(Bk[k], d0), d1, q);
            q = pk_fma(pk_mul(Ak[k], d0), d0, q);
            v2f p;
            p.x = fast_exp2(q.x);           // v_exp_f32 (trans pipe, co-executes)
            p.y = fast_exp2(q.y);
            r0 = pk_fma(p, W0[k], r0);
            r1 = pk_fma(p, W1[k], r1);
        }
        const v2f oe = pk_mul(x0, r0);      // channel 0 of both pairs
        const v2f oo = pk_mul(x1, r1);      // channel 1 of both pairs
        return (v4f){oe.x, oo.x, oe.y, oo.y};
    };

    const size_t base = (size_t)row0 * COLS + col;
    const v4f* __restrict__ xb = (const v4f*)x + base;
    v4f* __restrict__ ob = (v4f*)out + base;

    if (row0 + IPT <= rows) {
        // Fast path: all 8 row offsets are j*8192 bytes -> 24-bit ioffsets,
        // zero per-iteration address math; loads clause together.
#pragma unroll
        for (int j = 0; j < IPT; ++j) {
            ob[(size_t)j * COLS] = process(xb[(size_t)j * COLS]);
        }
    } else {
        for (int j = 0; j < IPT && (row0 + j) < rows; ++j) {
            ob[(size_t)j * COLS] = process(xb[(size_t)j * COLS]);
        }
    }
}

extern "C" void kernel_launch(void* const* d_in, const int* in_sizes, int n_in,
                              void* d_out, int out_size, void* d_ws, size_t ws_size,
                              hipStream_t stream) {
    const float* x     = (const float*)d_in[0];
    const float* means = (const float*)d_in[1];
    const float* icov  = (const float*)d_in[2];
    const float* wts   = (const float*)d_in[3];
    float* out = (float*)d_out;

    const int rows = in_sizes[0] / D_DIM;                 // B*S = 8192
    const int rowChunks = (rows + IPT - 1) / IPT;         // 1024
    const long threads = (long)rowChunks * COLS;          // 524288
    const int blocks = (int)((threads + 255) / 256);      // 2048 blocks, wave32 x8

    gmm2d_act_kernel<<<dim3(blocks), dim3(256), 0, stream>>>(x, means, icov, wts, out, rows);
}